// InteractionBlock_49847390437979
// MI455X (gfx1250) — compile-verified
//
#include <hip/hip_runtime.h>
#include <hip/hip_bf16.h>

#define N_ATOMS 50000
#define N_EDGES 800000
#define NF      128      // N_ATOM_BASIS == N_FILTERS == 128
#define NG      64       // N_GAUSS
#define CUTOFF  5.0f
#define EDGE_GROUPS 8    // 16-edge groups per block in edge_kernel

typedef __attribute__((ext_vector_type(2))) float v2f;
typedef __attribute__((ext_vector_type(8))) float v8f;

// D(16x16,f32) = A(16x4,f32) * B(4x16,f32) + C   — full-precision matrix pipe op.
// A layout (ISA 7.12.2): lanes 0-15 hold K=0,1 ; lanes 16-31 hold K=2,3 (2 VGPRs).
// B/C/D: row striped across lanes within a VGPR.
__device__ __forceinline__ v8f wmma_f32(v2f a, v2f b, v8f c) {
    return __builtin_amdgcn_wmma_f32_16x16x4_f32(
        /*neg_a=*/false, a, /*neg_b=*/false, b,
        /*c_mod=*/(short)0, c, /*reuse_a=*/false, /*reuse_b=*/false);
}

__device__ __forceinline__ float shifted_softplus(float x) {
    // softplus(x) - ln2, numerically stable
    float ax = fabsf(x);
    return fmaxf(x, 0.0f) + __logf(1.0f + __expf(-ax)) - 0.69314718056f;
}

__global__ void zero_kernel(float* __restrict__ p, int n) {
    int i = blockIdx.x * blockDim.x + threadIdx.x;
    if (i < n) p[i] = 0.0f;
}

// rf = r @ W_af  : [N_ATOMS x 128] x [128 x 128], 16 rows per block, 8 waves,
// each wave owns one 16-col tile, K stepped by 4 with f32 WMMA.
__global__ void rf_kernel(const float* __restrict__ r,
                          const float* __restrict__ W_af,
                          float* __restrict__ rf) {
    const int row0 = blockIdx.x * 16;
    const int t    = threadIdx.x;
    const int lane = t & 31;
    const int wave = t >> 5;            // 0..7
    const int l15  = lane & 15;
    const int lh   = lane >> 4;         // 0 or 1
    const int n    = wave * 16 + l15;   // output column

    v8f c = (v8f)0.0f;
    const float* arow = r + (size_t)(row0 + l15) * NF + 2 * lh;
#pragma unroll
    for (int k = 0; k < NF; k += 4) {
        v2f av = *(const v2f*)(arow + k);                 // 8B-aligned float2
        v2f bv;
        bv.x = W_af[(k + 2 * lh)     * NF + n];
        bv.y = W_af[(k + 2 * lh + 1) * NF + n];
        c = wmma_f32(av, bv, c);
    }
#pragma unroll
    for (int i = 0; i < 8; ++i)
        rf[(size_t)(row0 + i + 8 * lh) * NF + n] = c[i];
}

// Fused edge kernel: EDGE_GROUPS x 16 edges per block.
// The wave's 64x16 slice of W_df2 (the WMMA B operand) is hoisted into
// 16 float2 registers ONCE per block and reused across all groups — this
// removes ~1.4 GB of redundant weight traffic vs reloading per 16 edges.
// Per group:
//   g[16x64]  = exp(coeff*(e-off)^2)  -> LDS (row-padded to 66)
//   W[16x128] = g @ W_df2 + b_df2      (WMMA, bias-preloaded accumulators)
//   y         = rf[src] * W            (gather)
//   y_acc[dst] += y                    (hw fp32 atomics, no-return -> STOREcnt)
__global__ void edge_kernel(const float* __restrict__ e,
                            const int*   __restrict__ a,
                            const float* __restrict__ W_df2,
                            const float* __restrict__ b_df2,
                            const float* __restrict__ rf,
                            float* __restrict__ y_acc) {
    __shared__ float g_lds[16 * 66];
    __shared__ int   dst_lds[16];
    __shared__ int   src_lds[16];

    const int t    = threadIdx.x;
    const int lane = t & 31;
    const int wave = t >> 5;
    const int l15  = lane & 15;
    const int lh   = lane >> 4;
    const int n    = wave * 16 + l15;          // filter column owned by this lane

    // ---- hoist B operand (W_df2 slice) + bias into registers ----
    v2f barr[NG / 4];
#pragma unroll
    for (int j = 0; j < NG / 4; ++j) {
        barr[j].x = W_df2[(4 * j + 2 * lh)     * NF + n];
        barr[j].y = W_df2[(4 * j + 2 * lh + 1) * NF + n];
    }
    const float bias = b_df2[n];

    const float width = CUTOFF / (float)(NG - 1);
    const float coeff = -0.5f / (width * width);
    const int   gm  = t >> 4;                  // edge row this thread smears
    const int   gj0 = (t & 15) << 2;           // 4 gaussians per thread

    const int blockBase = blockIdx.x * (16 * EDGE_GROUPS);

    for (int grp = 0; grp < EDGE_GROUPS; ++grp) {
        const int e0 = blockBase + grp * 16;

        if (t < 32) {                          // stage [dst,src] pairs
            int m = t >> 1;
            int v = a[(size_t)(e0 + m) * 2 + (t & 1)];
            if (t & 1) src_lds[m] = v; else dst_lds[m] = v;
        }
        {   // Gaussian smearing: each thread computes 4 of the 16*64 values
            float ev = e[e0 + gm];
#pragma unroll
            for (int jj = 0; jj < 4; ++jj) {
                float d = ev - (float)(gj0 + jj) * width;
                g_lds[gm * 66 + gj0 + jj] = __expf(coeff * d * d);
            }
        }
        __syncthreads();
        if (t < 16)                            // warm the gathered rf rows
            __builtin_prefetch(rf + (size_t)src_lds[t] * NF, 0, 3);

        v8f c;
#pragma unroll
        for (int i = 0; i < 8; ++i) c[i] = bias;   // column-only bias, C layout

        const float* grow = &g_lds[l15 * 66 + 2 * lh];
#pragma unroll
        for (int j = 0; j < NG / 4; ++j) {
            v2f av = *(const v2f*)(grow + 4 * j);   // ds_load_b64, pad-66 conflict-free
            c = wmma_f32(av, barr[j], c);
        }
#pragma unroll
        for (int i = 0; i < 8; ++i) {
            int m = i + 8 * lh;                     // edge row of this C element
            float val = c[i] * rf[(size_t)src_lds[m] * NF + n];
            unsafeAtomicAdd(&y_acc[(size_t)dst_lds[m] * NF + n], val);
        }
        __syncthreads();                       // fence g_lds/idx reuse next group
    }
}

// out = ssp(y_acc @ W_d1 + b_d1) @ W_d2 + b_d2, fused: layer-1 result staged
// in LDS (padded row 130) as the A-operand of layer 2.
__global__ void mlp_kernel(const float* __restrict__ y_acc,
                           const float* __restrict__ W_d1,
                           const float* __restrict__ b_d1,
                           const float* __restrict__ W_d2,
                           const float* __restrict__ b_d2,
                           float* __restrict__ out) {
    __shared__ float t_lds[16 * 130];
    const int row0 = blockIdx.x * 16;
    const int t    = threadIdx.x;
    const int lane = t & 31;
    const int wave = t >> 5;
    const int l15  = lane & 15;
    const int lh   = lane >> 4;
    const int n    = wave * 16 + l15;

    // ---- layer 1 ----
    v8f c;
    float bias1 = b_d1[n];
#pragma unroll
    for (int i = 0; i < 8; ++i) c[i] = bias1;

    const float* arow = y_acc + (size_t)(row0 + l15) * NF + 2 * lh;
#pragma unroll
    for (int k = 0; k < NF; k += 4) {
        v2f av = *(const v2f*)(arow + k);
        v2f bv;
        bv.x = W_d1[(k + 2 * lh)     * NF + n];
        bv.y = W_d1[(k + 2 * lh + 1) * NF + n];
        c = wmma_f32(av, bv, c);
    }
#pragma unroll
    for (int i = 0; i < 8; ++i)
        t_lds[(i + 8 * lh) * 130 + n] = shifted_softplus(c[i]);
    __syncthreads();

    // ---- layer 2 (A from LDS) ----
    v8f c2;
    float bias2 = b_d2[n];
#pragma unroll
    for (int i = 0; i < 8; ++i) c2[i] = bias2;

    const float* trow = &t_lds[l15 * 130 + 2 * lh];
#pragma unroll
    for (int k = 0; k < NF; k += 4) {
        v2f av = *(const v2f*)(trow + k);       // ds_load_b64
        v2f bv;
        bv.x = W_d2[(k + 2 * lh)     * NF + n];
        bv.y = W_d2[(k + 2 * lh + 1) * NF + n];
        c2 = wmma_f32(av, bv, c2);
    }
#pragma unroll
    for (int i = 0; i < 8; ++i)
        out[(size_t)(row0 + i + 8 * lh) * NF + n] = c2[i];
}

extern "C" void kernel_launch(void* const* d_in, const int* in_sizes, int n_in,
                              void* d_out, int out_size, void* d_ws, size_t ws_size,
                              hipStream_t stream) {
    // setup_inputs order: r, e, a, W_df1, b_df1, W_df2, b_df2, W_af, W_d1, b_d1, W_d2, b_d2
    const float* r     = (const float*)d_in[0];
    const float* e     = (const float*)d_in[1];
    const int*   a     = (const int*)  d_in[2];
    // d_in[3], d_in[4] (W_df1/b_df1) are dead code in the reference — skipped.
    const float* W_df2 = (const float*)d_in[5];
    const float* b_df2 = (const float*)d_in[6];
    const float* W_af  = (const float*)d_in[7];
    const float* W_d1  = (const float*)d_in[8];
    const float* b_d1  = (const float*)d_in[9];
    const float* W_d2  = (const float*)d_in[10];
    const float* b_d2  = (const float*)d_in[11];
    float* out = (float*)d_out;

    float* rf    = (float*)d_ws;                       // 50000*128 floats
    float* y_acc = rf + (size_t)N_ATOMS * NF;          // 50000*128 floats

    const int acc_elems = N_ATOMS * NF;                // 6,400,000
    zero_kernel<<<acc_elems / 256, 256, 0, stream>>>(y_acc, acc_elems);
    rf_kernel  <<<N_ATOMS / 16, 256, 0, stream>>>(r, W_af, rf);
    edge_kernel<<<N_EDGES / (16 * EDGE_GROUPS), 256, 0, stream>>>(
        e, a, W_df2, b_df2, rf, y_acc);
    mlp_kernel <<<N_ATOMS / 16, 256, 0, stream>>>(y_acc, W_d1, b_d1, W_d2, b_d2, out);
}